// ResGCN_15195594293931
// MI455X (gfx1250) — compile-verified
//
#include <hip/hip_runtime.h>

typedef float v2f __attribute__((ext_vector_type(2)));
typedef float v8f __attribute__((ext_vector_type(8)));

#define NNODES   50000
#define NHID     128
#define NCLASSES 40
#define MTILES   (NNODES / 16)   // 3125, exact

// ---------------------------------------------------------------------------
// C[M,128] = A[M,128] @ W[128,128]   (fp32 WMMA 16x16x4)
// 1 block = one 16-row M tile; 8 waves = 8 N tiles of 16 cols.
// A tile staged in LDS with stride 132 (4-float pad) -> conflict-free b64 reads.
// ---------------------------------------------------------------------------
__global__ __launch_bounds__(256) void gemm128_wmma(
    const float* __restrict__ A, const float* __restrict__ W,
    float* __restrict__ C)
{
    __shared__ float Alds[16 * 132];
    const int tid   = threadIdx.x;
    const int tileM = blockIdx.x;

    // cooperative load: 16x128 floats = 512 float4, 256 threads x 2
#pragma unroll
    for (int j = 0; j < 2; ++j) {
        const int idx = tid + j * 256;        // float4 slot
        const int row = idx >> 5;             // 32 float4 per row
        const int col = (idx & 31) << 2;
        const float4 v = *(const float4*)(A + (size_t)(tileM * 16 + row) * NHID + col);
        float* d = &Alds[row * 132 + col];
        d[0] = v.x; d[1] = v.y; d[2] = v.z; d[3] = v.w;
    }
    __syncthreads();

    const int lane  = tid & 31;
    const int waveN = tid >> 5;               // N tile 0..7
    const int r16   = lane & 15;
    const int kb    = (lane >> 4) << 1;       // 0 (lanes 0-15) or 2 (lanes 16-31)
    const int col   = (waveN << 4) + r16;

    v8f acc = {};
    const float* Ar = &Alds[r16 * 132 + kb];
    const float* Wc = W + col;
#pragma unroll
    for (int k = 0; k < NHID; k += 4) {
        v2f a; a.x = Ar[k];                       a.y = Ar[k + 1];
        v2f b; b.x = Wc[(size_t)(k + kb) * NHID]; b.y = Wc[(size_t)(k + kb + 1) * NHID];
        acc = __builtin_amdgcn_wmma_f32_16x16x4_f32(
            false, a, false, b, (short)0, acc, false, false);
    }

    const int rowBase = tileM * 16 + ((lane >> 4) << 3);   // +8 for upper half-wave
#pragma unroll
    for (int r = 0; r < 8; ++r)
        C[(size_t)(rowBase + r) * NHID + col] = acc[r];
}

// ---------------------------------------------------------------------------
// Out[M,40] = H[M,128] @ Wout[128,40] + bout   (N padded to 48 = 3 WMMA tiles)
// ---------------------------------------------------------------------------
__global__ __launch_bounds__(96) void gemm_out_wmma(
    const float* __restrict__ H, const float* __restrict__ Wout,
    const float* __restrict__ bout, float* __restrict__ Out)
{
    __shared__ float Wl[128 * 48];
    __shared__ float bl[48];
    const int tid = threadIdx.x;

    for (int i = tid; i < 128 * 48; i += 96) {
        const int r = i / 48, c = i - r * 48;
        Wl[i] = (c < NCLASSES) ? Wout[r * NCLASSES + c] : 0.0f;
    }
    if (tid < 48) bl[tid] = (tid < NCLASSES) ? bout[tid] : 0.0f;
    __syncthreads();

    const int lane  = tid & 31;
    const int waveN = tid >> 5;               // 0..2
    const int r16   = lane & 15;
    const int kb    = (lane >> 4) << 1;
    const int col   = waveN * 16 + r16;
    const int tileM = blockIdx.x;

    const float* Hr = H + (size_t)(tileM * 16 + r16) * NHID + kb;
    v8f acc = {};
#pragma unroll
    for (int k = 0; k < NHID; k += 4) {
        const v2f a = *(const v2f*)(Hr + k);
        v2f b; b.x = Wl[(k + kb) * 48 + col]; b.y = Wl[(k + kb + 1) * 48 + col];
        acc = __builtin_amdgcn_wmma_f32_16x16x4_f32(
            false, a, false, b, (short)0, acc, false, false);
    }

    if (col < NCLASSES) {                      // predication only after all WMMAs
        const float bb = bl[col];
        const int rowBase = tileM * 16 + ((lane >> 4) << 3);
#pragma unroll
        for (int r = 0; r < 8; ++r)
            Out[(size_t)(rowBase + r) * NCLASSES + col] = acc[r] + bb;
    }
}

// ---------------------------------------------------------------------------
// AGG[dst] += w * T[src]  : wave per edge, float4 gather, f32 HW atomics
// ---------------------------------------------------------------------------
__global__ __launch_bounds__(256) void spmm_scatter(
    const int* __restrict__ esrc, const int* __restrict__ edst,
    const float* __restrict__ ew, const float* __restrict__ T,
    float* __restrict__ AGG, int nE)
{
    const int lane   = threadIdx.x & 31;
    const int waves  = (gridDim.x * blockDim.x) >> 5;
    int e = (blockIdx.x * blockDim.x + threadIdx.x) >> 5;
    for (; e < nE; e += waves) {
        const int   s = esrc[e];
        const int   d = edst[e];
        const float w = ew[e];
        const float4 v = *(const float4*)(T + (size_t)s * NHID + lane * 4);
        float* dst = AGG + (size_t)d * NHID + lane * 4;
        unsafeAtomicAdd(dst + 0, w * v.x);   // global_atomic_add_f32 (no return)
        unsafeAtomicAdd(dst + 1, w * v.y);
        unsafeAtomicAdd(dst + 2, w * v.z);
        unsafeAtomicAdd(dst + 3, w * v.w);
    }
}

// ---------------------------------------------------------------------------
// H = relu(AGG + bias) [+ residual]
// ---------------------------------------------------------------------------
__global__ __launch_bounds__(256) void bias_relu_res(
    const float* __restrict__ agg, const float* __restrict__ bias,
    const float* __restrict__ res, float* __restrict__ H, int n4)
{
    const int i = blockIdx.x * blockDim.x + threadIdx.x;
    if (i >= n4) return;
    const int cb = (i & 31) << 2;             // column base within 128
    float4 a = ((const float4*)agg)[i];
    a.x = fmaxf(a.x + bias[cb + 0], 0.0f);
    a.y = fmaxf(a.y + bias[cb + 1], 0.0f);
    a.z = fmaxf(a.z + bias[cb + 2], 0.0f);
    a.w = fmaxf(a.w + bias[cb + 3], 0.0f);
    if (res != nullptr) {
        const float4 r = ((const float4*)res)[i];
        a.x += r.x; a.y += r.y; a.z += r.z; a.w += r.w;
    }
    ((float4*)H)[i] = a;
}

__global__ __launch_bounds__(256) void zero_f32(float4* __restrict__ p, int n4)
{
    const int i = blockIdx.x * blockDim.x + threadIdx.x;
    if (i < n4) p[i] = make_float4(0.f, 0.f, 0.f, 0.f);
}

// ---------------------------------------------------------------------------
extern "C" void kernel_launch(void* const* d_in, const int* in_sizes, int n_in,
                              void* d_out, int out_size, void* d_ws, size_t ws_size,
                              hipStream_t stream)
{
    const float* x    = (const float*)d_in[0];
    const int*   esrc = (const int*)  d_in[1];
    const int*   edst = (const int*)  d_in[2];
    const float* ew   = (const float*)d_in[3];
    const float* W0   = (const float*)d_in[4];
    const float* b0   = (const float*)d_in[5];
    const float* Ws   = (const float*)d_in[6];   // [3][128][128]
    const float* bs   = (const float*)d_in[7];   // [3][128]
    const float* Wout = (const float*)d_in[8];
    const float* bout = (const float*)d_in[9];
    float*       out  = (float*)d_out;
    const int    nE   = in_sizes[1];

    const size_t nFeat = (size_t)NNODES * NHID;  // 6.4M floats
    float* T   = (float*)d_ws;
    float* AGG = T + nFeat;
    float* H   = AGG + nFeat;

    const int n4      = (int)(nFeat / 4);
    const int gElem   = (n4 + 255) / 256;
    const dim3 gGemm(MTILES);
    const dim3 gSpmm(16384);                     // 131072 waves, grid-stride

    // layer 0: T = x @ W0 ; AGG = A*T ; H = relu(AGG + b0)
    gemm128_wmma<<<gGemm, 256, 0, stream>>>(x, W0, T);
    zero_f32<<<gElem, 256, 0, stream>>>((float4*)AGG, n4);
    spmm_scatter<<<gSpmm, 256, 0, stream>>>(esrc, edst, ew, T, AGG, nE);
    bias_relu_res<<<gElem, 256, 0, stream>>>(AGG, b0, nullptr, H, n4);

    // residual layers 1..3
    for (int l = 0; l < 3; ++l) {
        gemm128_wmma<<<gGemm, 256, 0, stream>>>(H, Ws + (size_t)l * NHID * NHID, T);
        zero_f32<<<gElem, 256, 0, stream>>>((float4*)AGG, n4);
        spmm_scatter<<<gSpmm, 256, 0, stream>>>(esrc, edst, ew, T, AGG, nE);
        bias_relu_res<<<gElem, 256, 0, stream>>>(AGG, bs + (size_t)l * NHID, H, H, n4);
    }

    // output projection
    gemm_out_wmma<<<gGemm, 96, 0, stream>>>(H, Wout, bout, out);
}